// ProjectionBlock_15444702397274
// MI455X (gfx1250) — compile-verified
//
#include <hip/hip_runtime.h>
#include <math.h>

// Problem constants (from reference)
#define NUM_MAPS 16
#define D_IN     512
#define D_OUT    128
#define Q_DIM    64
#define BATCH    256

// LDS tiling
#define KC      32           // K-chunk for the main GEMM
#define SW_STR  36           // sW row stride (KC + 4 pad)
#define SXP_STR 132          // sXp paired-row stride (2*Q_DIM + 4 pad)
#define SYP_STR 132          // sYp paired-row stride (2*Q_DIM + 4 pad)
#define SG_STR  68           // sG row stride (Q_DIM + 4 pad)

typedef __attribute__((ext_vector_type(2))) float v2f;
typedef __attribute__((ext_vector_type(8))) float v8f;

// K-pair-interleaved layout: element (k, n) lives at  [k>>1]*STRIDE + 2*n + (k&1).
// A WMMA f32 16x16x4 fragment needs, per lane, values at (k, c) and (k+1, c) with
// k even -> adjacent floats -> one ds_load_b64 straight into an even VGPR pair.

// One workgroup (256 threads = 8 wave32) per (map, batch) pair.
//  Phase A: Y = W_m @ X_b            (128x512 @ 512x64, f32 WMMA, Y -> LDS paired)
//  Phase B: G = Y^T Y                (64x64, f32 WMMA from LDS)
//  Phase C: Cholesky  G = R^T R      (cooperative, in LDS)
//  Phase D: Q = Y R^-1               (independent per-row forward substitution)
//  Phase E: write Q (row-major) to global
__launch_bounds__(256)
__global__ void frmap_choleskyqr_kernel(const float* __restrict__ X,
                                        const float* __restrict__ W,
                                        float* __restrict__ out) {
    // LDS layout (floats):
    //   sYp : 64*132 = 8448           persistent (Y, K-pair interleaved over rows)
    //   union:
    //     sW  : 128*36 = 4608 | sXp : 16*132 = 2112   (Phase A staging)
    //     sG  : 64*68  = 4352 | sRinv : 64            (Phases B-D)
    __shared__ __align__(16) float smem[8448 + 6720];
    float* sYp   = smem;                      // 64 x SYP_STR (Y rows, paired)
    float* sW    = smem + 64 * SYP_STR;       // 128 x SW_STR
    float* sXp   = sW + 128 * SW_STR;         // 16 x SXP_STR (X chunk rows, paired)
    float* sG    = smem + 64 * SYP_STR;       // 64 x SG_STR (reuses sW/sXp space)
    float* sRinv = sG + 64 * SG_STR;          // 64

    const int bid  = blockIdx.x;
    const int m    = bid / BATCH;
    const int b    = bid - m * BATCH;
    const float* Wm = W + (size_t)m * (D_OUT * D_IN);
    const float* Xb = X + (size_t)b * (D_IN * Q_DIM);

    const int tid  = threadIdx.x;
    const int wave = tid >> 5;
    const int lane = tid & 31;
    const int half = lane >> 4;   // 0: lanes 0-15, 1: lanes 16-31
    const int l16  = lane & 15;

    // ---------------- Phase A: Y = W_m @ X_b ----------------
    // wave w owns output rows [16w, 16w+16), all 4 N-tiles of 16 cols.
    v8f acc0 = {}, acc1 = {}, acc2 = {}, acc3 = {};

    for (int c = 0; c < D_IN / KC; ++c) {
        const int k0 = c * KC;
        // stage W chunk: 128 rows x KC cols (float4-coalesced, row-major)
        #pragma unroll
        for (int r = 0; r < 4; ++r) {
            const int t = tid + r * 256;          // 0..1023
            const int o = t >> 3, f = t & 7;      // row, float4-slot
            const float4 v = *reinterpret_cast<const float4*>(Wm + o * D_IN + k0 + 4 * f);
            *reinterpret_cast<float4*>(sW + o * SW_STR + 4 * f) = v;
        }
        // stage X chunk: KC rows x 64 cols, scattered into K-pair-interleaved layout
        #pragma unroll
        for (int r = 0; r < 2; ++r) {
            const int t = tid + r * 256;          // 0..511
            const int kk = t >> 4, f = t & 15;    // chunk-row, float4-slot (n0 = 4f)
            const float4 v = *reinterpret_cast<const float4*>(Xb + (size_t)(k0 + kk) * Q_DIM + 4 * f);
            float* dst = sXp + (kk >> 1) * SXP_STR + (kk & 1) + 8 * f;
            dst[0] = v.x; dst[2] = v.y; dst[4] = v.z; dst[6] = v.w;
        }
        __syncthreads();

        // A-fragment (16x4 f32): lane l, vgpr p -> A[M = l&15][K = 2*(l>>4)+p]
        const float* aRow = sW + (16 * wave + l16) * SW_STR + 2 * half;
        #pragma unroll
        for (int kk = 0; kk < KC; kk += 4) {
            v2f A;
            A.x = aRow[kk + 0];
            A.y = aRow[kk + 1];
            // B-fragment (4x16 f32): lane l, vgpr p -> B[K = 2*(l>>4)+p][N = l&15]
            // K = kk + 2*half (even): pair row (kk>>1)+half, adjacent floats.
            const float* bBase = sXp + ((kk >> 1) + half) * SXP_STR + 2 * l16;
            const v2f B0 = *reinterpret_cast<const v2f*>(bBase +  0);
            const v2f B1 = *reinterpret_cast<const v2f*>(bBase + 32);
            const v2f B2 = *reinterpret_cast<const v2f*>(bBase + 64);
            const v2f B3 = *reinterpret_cast<const v2f*>(bBase + 96);
            acc0 = __builtin_amdgcn_wmma_f32_16x16x4_f32(false, A, false, B0, (short)0, acc0, false, false);
            acc1 = __builtin_amdgcn_wmma_f32_16x16x4_f32(false, A, false, B1, (short)0, acc1, false, false);
            acc2 = __builtin_amdgcn_wmma_f32_16x16x4_f32(false, A, false, B2, (short)0, acc2, false, false);
            acc3 = __builtin_amdgcn_wmma_f32_16x16x4_f32(false, A, false, B3, (short)0, acc3, false, false);
        }
        __syncthreads();
    }

    // C/D 16x16 f32 layout: lane l, vgpr v holds [M = v + 8*(l>>4)][N = l&15].
    // Store Y row-pair interleaved: Y[M][N] -> sYp[M>>1][2N + (M&1)].
    #pragma unroll
    for (int v = 0; v < 8; ++v) {
        const int M = 16 * wave + v + 8 * half;
        float* dst = sYp + (M >> 1) * SYP_STR + (M & 1) + 2 * l16;
        dst[  0] = acc0[v];
        dst[ 32] = acc1[v];
        dst[ 64] = acc2[v];
        dst[ 96] = acc3[v];
    }
    __syncthreads();

    // ---------------- Phase B: G = Y^T Y (64x64) ----------------
    // 16 tiles of 16x16; wave w owns tiles t = 2w, 2w+1 (same row-block gi).
    // Fragment element (q, kidx) = Y[kidx][q] -> sYp[kidx>>1][2q + (kidx&1)];
    // kidx = k + 2*half (even) and kidx+1 are adjacent -> single b64 load.
    {
        const int t0  = 2 * wave;
        const int gi  = t0 >> 2;        // G row block
        const int gj0 = t0 & 3;         // G col block (0 or 2); gj1 = gj0+1
        v8f g0 = {}, g1 = {};
        #pragma unroll
        for (int k = 0; k < D_OUT; k += 4) {
            const float* base = sYp + ((k >> 1) + half) * SYP_STR;
            const v2f A  = *reinterpret_cast<const v2f*>(base + 2 * (16 * gi + l16));
            const v2f B0 = *reinterpret_cast<const v2f*>(base + 2 * (16 * gj0 + l16));
            const v2f B1 = *reinterpret_cast<const v2f*>(base + 2 * (16 * (gj0 + 1) + l16));
            g0 = __builtin_amdgcn_wmma_f32_16x16x4_f32(false, A, false, B0, (short)0, g0, false, false);
            g1 = __builtin_amdgcn_wmma_f32_16x16x4_f32(false, A, false, B1, (short)0, g1, false, false);
        }
        #pragma unroll
        for (int v = 0; v < 8; ++v) {
            const int row = (16 * gi + v + 8 * half) * SG_STR + l16;
            sG[row + 16 * gj0]       = g0[v];
            sG[row + 16 * (gj0 + 1)] = g1[v];
        }
    }
    __syncthreads();

    // ---------------- Phase C: Cholesky G = R^T R (upper R in sG) ----------------
    for (int j = 0; j < Q_DIM; ++j) {
        if (tid == 0) {
            const float d = sqrtf(sG[j * SG_STR + j]);
            sG[j * SG_STR + j] = d;
            sRinv[j] = 1.0f / d;
        }
        __syncthreads();
        const float rinv = sRinv[j];
        for (int k = j + 1 + tid; k < Q_DIM; k += 256)
            sG[j * SG_STR + k] *= rinv;
        __syncthreads();
        const int n = Q_DIM - 1 - j;
        const int total = n * n;
        for (int t = tid; t < total; t += 256) {
            const int i  = j + 1 + t / n;
            const int kk = j + 1 + t % n;
            sG[i * SG_STR + kk] -= sG[j * SG_STR + i] * sG[j * SG_STR + kk];
        }
        __syncthreads();
    }

    // ---------------- Phase D: Q = Y R^-1 (per-row forward substitution) ----------------
    // Row i only touches its own sYp row slots -> fully independent, no barriers.
    // Y[i][j] lives at sYp[i>>1][2j + (i&1)] (stride-2, conflict-free across lanes).
    if (tid < D_OUT) {
        float* yr = sYp + (tid >> 1) * SYP_STR + (tid & 1);
        for (int j = 0; j < Q_DIM; ++j) {
            float a = yr[2 * j];
            for (int k = 0; k < j; ++k)
                a -= yr[2 * k] * sG[k * SG_STR + j];
            yr[2 * j] = a * sRinv[j];
        }
    }
    __syncthreads();

    // ---------------- Phase E: write Q ----------------
    float* outp = out + (size_t)bid * (D_OUT * Q_DIM);
    for (int t = tid; t < D_OUT * Q_DIM; t += 256) {
        const int row = t >> 6, col = t & 63;
        outp[t] = sYp[(row >> 1) * SYP_STR + 2 * col + (row & 1)];
    }
}

extern "C" void kernel_launch(void* const* d_in, const int* in_sizes, int n_in,
                              void* d_out, int out_size, void* d_ws, size_t ws_size,
                              hipStream_t stream) {
    const float* X = (const float*)d_in[0];   // (BATCH, D_IN, Q_DIM)
    const float* W = (const float*)d_in[1];   // (NUM_MAPS, D_OUT, D_IN)
    float* out = (float*)d_out;               // (NUM_MAPS*BATCH, D_OUT, Q_DIM)
    (void)in_sizes; (void)n_in; (void)out_size; (void)d_ws; (void)ws_size;

    dim3 grid(NUM_MAPS * BATCH);              // 4096 workgroups, one per (m, b)
    dim3 block(256);                          // 8 wave32
    frmap_choleskyqr_kernel<<<grid, block, 0, stream>>>(X, W, out);
}